// Loss_2_8323646620406
// MI455X (gfx1250) — compile-verified
//
#include <hip/hip_runtime.h>

// Problem constants from the reference
#define BB 64
#define SS 16384
#define CC 20
#define NTOT (BB * SS)            // 1,048,576 elements
#define W0c 0.51f
#define W1c 19.05f

#define NBLOCKS 512
#define BS 256

typedef __attribute__((ext_vector_type(2))) float v2f;
typedef __attribute__((ext_vector_type(8))) float v8f;

// ---------------------------------------------------------------------------
// Pass 1: grid-stride over vec4 chunks; per-element loss; LDS tree reduction
// to one partial per block (deterministic). Gather from y_pred_comb is
// predicated on ys==1, so ~half the 80 MiB gather traffic is never fetched.
// ---------------------------------------------------------------------------
__global__ void loss_partial_kernel(const float4* __restrict__ ps4,
                                    const float* __restrict__ comb,
                                    const float4* __restrict__ ys4,
                                    const int4* __restrict__ idx4,
                                    float* __restrict__ partials,
                                    int nvec) {
    __shared__ float red[BS];
    float acc = 0.0f;

    const int stride = gridDim.x * blockDim.x;
    for (int i = blockIdx.x * blockDim.x + threadIdx.x; i < nvec; i += stride) {
        float4 p = ps4[i];
        float4 y = ys4[i];
        int4   c = idx4[i];
        // base element index of this vec4 chunk
        const size_t base = (size_t)i * 4u * (size_t)CC;

        // element 0
        if (y.x > 0.5f) {
            float g = comb[base + 0 * CC + (size_t)c.x];
            acc += -W1c * __logf(p.x) - __logf(g);
        } else {
            acc += -W0c * __logf(1.0f - p.x);
        }
        // element 1
        if (y.y > 0.5f) {
            float g = comb[base + 1 * CC + (size_t)c.y];
            acc += -W1c * __logf(p.y) - __logf(g);
        } else {
            acc += -W0c * __logf(1.0f - p.y);
        }
        // element 2
        if (y.z > 0.5f) {
            float g = comb[base + 2 * CC + (size_t)c.z];
            acc += -W1c * __logf(p.z) - __logf(g);
        } else {
            acc += -W0c * __logf(1.0f - p.z);
        }
        // element 3
        if (y.w > 0.5f) {
            float g = comb[base + 3 * CC + (size_t)c.w];
            acc += -W1c * __logf(p.w) - __logf(g);
        } else {
            acc += -W0c * __logf(1.0f - p.w);
        }
    }

    // deterministic block tree reduction in LDS
    const int tid = threadIdx.x;
    red[tid] = acc;
    __syncthreads();
#pragma unroll
    for (int s = BS / 2; s > 0; s >>= 1) {
        if (tid < s) red[tid] += red[tid + s];
        __syncthreads();
    }
    if (tid == 0) partials[blockIdx.x] = red[0];
}

// ---------------------------------------------------------------------------
// Pass 2: single wave32. Each lane sums its strided slice of the per-block
// partials, then a V_WMMA_F32_16X16X4_F32 with B = ones performs the
// cross-lane reduction on the matrix pipe:
//   A (16x4): VGPR0 = lane partials (k=0 lanes 0-15, k=2 lanes 16-31),
//             VGPR1 = 0 (k=1,3)  =>  D[m][n] = s_m + s_{m+16}  for all n.
// Column 0 lives in lane 0 (M=0..7 across the 8 D VGPRs) and lane 16
// (M=8..15), so total = sum(D vgprs)@lane0 + sum(D vgprs)@lane16.
// All 32 lanes active, no divergence around the WMMA (EXEC == all ones).
// ---------------------------------------------------------------------------
__global__ void loss_final_kernel(const float* __restrict__ partials,
                                  int nb,
                                  float* __restrict__ out) {
    const int lane = threadIdx.x;  // 0..31
    float s = 0.0f;
    for (int i = lane; i < nb; i += 32) s += partials[i];

    v2f a;
    a.x = s;
    a.y = 0.0f;
    v2f b;
    b.x = 1.0f;
    b.y = 1.0f;
    v8f c = {};
    v8f d = __builtin_amdgcn_wmma_f32_16x16x4_f32(
        /*neg_a=*/false, a, /*neg_b=*/false, b,
        /*c_mod=*/(short)0, c, /*reuse_a=*/false, /*reuse_b=*/false);

    float t = ((d[0] + d[1]) + (d[2] + d[3])) + ((d[4] + d[5]) + (d[6] + d[7]));
    float total = __shfl(t, 0, 32) + __shfl(t, 16, 32);

    if (lane == 0) out[0] = total * (1.0f / (float)NTOT);
}

extern "C" void kernel_launch(void* const* d_in, const int* in_sizes, int n_in,
                              void* d_out, int out_size, void* d_ws, size_t ws_size,
                              hipStream_t stream) {
    // setup_inputs() order: y_pred_stroke, y_pred_comb, y_stroke, y_comb
    const float4* ps4  = (const float4*)d_in[0];
    const float*  comb = (const float*)d_in[1];
    const float4* ys4  = (const float4*)d_in[2];
    const int4*   idx4 = (const int4*)d_in[3];
    float* out = (float*)d_out;

    float* partials = (float*)d_ws;  // NBLOCKS floats = 2 KiB scratch

    const int nvec = NTOT / 4;
    loss_partial_kernel<<<NBLOCKS, BS, 0, stream>>>(ps4, comb, ys4, idx4, partials, nvec);
    loss_final_kernel<<<1, 32, 0, stream>>>(partials, NBLOCKS, out);
}